// CTCLayer_76527727280268
// MI455X (gfx1250) — compile-verified
//
#include <hip/hip_runtime.h>
#include <math.h>

#define BB 512
#define TT 256
#define CC 100
#define LL 64
#define SS 129            // 2*LL + 1
#define BLANKC 99         // C - 1
#define NEG_INF_F (-1e30f)
#define EPS_F (1e-7f)
#define TCHUNK 32
#define NCHUNK 8          // TT / TCHUNK
#define NTHREADS 160      // 5 waves of 32; 129 active state threads

typedef unsigned int u32x4 __attribute__((ext_vector_type(4)));
typedef int i32x8 __attribute__((ext_vector_type(8)));
typedef int i32x4 __attribute__((ext_vector_type(4)));

// Issue a TDM 2-D tile load: rows x rowlen f32 elements, contiguous rows
// (tensor_dim0_stride == rowlen), from global 'gsrc' into LDS byte offset
// 'lds_byte_off'. Descriptor bit layout per CDNA5 ISA 8.3/8.4.
__device__ __forceinline__ void tdm_load_rows(unsigned lds_byte_off,
                                              const float* gsrc,
                                              unsigned rows, unsigned rowlen) {
#if defined(__HIP_DEVICE_COMPILE__)
  unsigned long long ga = (unsigned long long)(size_t)gsrc;
  u32x4 g0;
  g0.x = 1u;                                        // count=1, user descriptor
  g0.y = lds_byte_off;                              // lds_addr [63:32]
  g0.z = (unsigned)(ga & 0xFFFFFFFFu);              // global_addr[31:0]
  g0.w = (unsigned)((ga >> 32) & 0x01FFFFFFu)       // global_addr[56:32]
       | (2u << 30);                                // type=2 ("image")
  i32x8 g1;
  g1[0] = (int)(2u << 16);                          // data_size=2 -> 4 bytes
  g1[1] = (int)(rowlen << 16);                      // tensor_dim0[15:0] @ bits 63:48
  g1[2] = (int)(rows << 16);                        // tensor_dim1[15:0] @ bits 95:80
  g1[3] = (int)(rowlen << 16);                      // tile_dim0 @ bits 127:112
  g1[4] = (int)rows;                                // tile_dim1 @ bits 143:128 (tile_dim2=0)
  g1[5] = (int)rowlen;                              // tensor_dim0_stride[31:0] @ bits 191:160
  g1[6] = 0;
  g1[7] = 0;
  i32x4 z4 = {0, 0, 0, 0};                          // groups 2/3 unused (2-D tensor)
  i32x8 z8 = {0, 0, 0, 0, 0, 0, 0, 0};              // extra group (clang-23 6-arg form)
  __builtin_amdgcn_tensor_load_to_lds(g0, g1, z4, z4, z8, 0);
#endif
}

__device__ __forceinline__ void tdm_wait_all() {
#if defined(__HIP_DEVICE_COMPILE__)
  __builtin_amdgcn_s_wait_tensorcnt(0);
#endif
}

__global__ __launch_bounds__(NTHREADS)
void ctc_alpha_kernel(const int* __restrict__ y_true,
                      const float* __restrict__ y_pred,
                      float* __restrict__ out) {
  // Double-buffered probability tiles staged by the TDM + double-buffered alpha.
  __shared__ float lpbuf[2][TCHUNK * CC];   // 2 x 12.8 KB
  __shared__ float alpha[2][SS + 3];

  const int b = blockIdx.x;
  const int s = threadIdx.x;
  const bool active = (s < SS);
  const bool wave0 = ((threadIdx.x >> 5) == 0);

  // Per-state extended label + skip-connection flag.
  int myLabel = BLANKC;
  bool mySkip = false;
  if (active && (s & 1)) {
    const int li = (s - 1) >> 1;
    myLabel = y_true[(size_t)b * LL + li];
    if (s >= 3) {
      const int prevLab = y_true[(size_t)b * LL + li - 1];
      mySkip = (myLabel != BLANKC) && (myLabel != prevLab);
    }
  }

  const float* base = y_pred + (size_t)b * TT * CC;
  const unsigned lp_off[2] = {
    (unsigned)(size_t)(&lpbuf[0][0]),   // generic LDS addr: low 32 bits = LDS offset
    (unsigned)(size_t)(&lpbuf[1][0])
  };

  // Prologue: DMA chunk 0 into buffer 0.
  if (wave0) {
    tdm_load_rows(lp_off[0], base, TCHUNK, CC);
    tdm_wait_all();
  }
  __syncthreads();

  // t = 0 initialization: alpha0[0]=lp(0,0), alpha0[1]=lp(0,1), else -inf.
  if (active) {
    const float v = logf(lpbuf[0][myLabel] + EPS_F);
    alpha[0][s] = (s < 2) ? v : NEG_INF_F;
  }
  __syncthreads();

  int cur = 0;   // alpha buffer holding step t's values
  int cb = 0;    // lp tile buffer for current chunk
  for (int c = 0; c < NCHUNK; ++c) {
    // Overlap: DMA next chunk into the other tile buffer while computing.
    if (wave0 && (c + 1 < NCHUNK)) {
      tdm_load_rows(lp_off[cb ^ 1], base + (size_t)(c + 1) * TCHUNK * CC,
                    TCHUNK, CC);
    }
    const float* lprow = &lpbuf[cb][0];
    const int tl0 = (c == 0) ? 1 : 0;     // t=0 already consumed
    for (int tl = tl0; tl < TCHUNK; ++tl) {
      if (active) {
        const float lpv = logf(lprow[tl * CC + myLabel] + EPS_F);
        const float a1 = alpha[cur][s];
        const float a2 = (s >= 1) ? alpha[cur][s - 1] : NEG_INF_F;
        const float a3 = mySkip ? alpha[cur][s - 2] : NEG_INF_F;
        const float m = fmaxf(a1, fmaxf(a2, a3));
        const float sum = expf(a1 - m) + expf(a2 - m) + expf(a3 - m);
        alpha[cur ^ 1][s] = m + logf(sum) + lpv;
      }
      __syncthreads();                    // one split-barrier per time step
      cur ^= 1;
    }
    // Drain the in-flight DMA before flipping tile buffers.
    if (wave0 && (c + 1 < NCHUNK)) tdm_wait_all();
    __syncthreads();
    cb ^= 1;
  }

  // -log likelihood = -logaddexp(alpha_T[S-1], alpha_T[S-2])
  if (threadIdx.x == 0) {
    const float aA = alpha[cur][SS - 1];
    const float aB = alpha[cur][SS - 2];
    const float m = fmaxf(aA, aB);
    out[b] = -(m + logf(expf(aA - m) + expf(aB - m)));
  }
}

extern "C" void kernel_launch(void* const* d_in, const int* in_sizes, int n_in,
                              void* d_out, int out_size, void* d_ws, size_t ws_size,
                              hipStream_t stream) {
  (void)in_sizes; (void)n_in; (void)d_ws; (void)ws_size; (void)out_size;
  const int*   y_true = (const int*)d_in[0];
  const float* y_pred = (const float*)d_in[1];
  float*       out    = (float*)d_out;
  ctc_alpha_kernel<<<dim3(BB), dim3(NTHREADS), 0, stream>>>(y_true, y_pred, out);
}